// OutputLayer_77283641524591
// MI455X (gfx1250) — compile-verified
//
#include <hip/hip_runtime.h>
#include <math.h>
#include <stdint.h>

typedef __attribute__((ext_vector_type(2))) float v2f;
typedef __attribute__((ext_vector_type(4))) float v4f;
typedef __attribute__((ext_vector_type(8))) float v8f;

#define SROWS 8192
#define DCOLS 4096
#define KSLICES 2
#define DHALF (DCOLS / KSLICES)          // 2048 columns per K-slice
#define WAVES_PER_BLOCK 4
#define BLOCK_THREADS (WAVES_PER_BLOCK * 32)               // 128
#define ROWS_PER_WAVE 16
#define ROWS_PER_BLOCK (WAVES_PER_BLOCK * ROWS_PER_WAVE)   // 64
#define KCHUNK 64
#define NCHUNK (DHALF / KCHUNK)          // 32
#define LDS_STRIDE 66                    // 64 + 2 pad -> conflict-light ds reads

// GEMV via V_WMMA_F32_16X16X4_F32 with double-buffered async global->LDS copies.
// Each wave computes logits for 16 rows over one K-slice of 2048 columns.
// A (16x4) = G tile staged through LDS; B (4x16) = w chunk broadcast across N.
// D[m][n] is identical across n == partial dot product of row m.
__global__ __launch_bounds__(BLOCK_THREADS) void gemv_wmma_kernel(
    const float* __restrict__ G, const float* __restrict__ w,
    float* __restrict__ logits_partial /* [KSLICES][SROWS] */) {
  __shared__ float sw[DHALF];                                             // 8 KB
  __shared__ float tile[2][WAVES_PER_BLOCK][ROWS_PER_WAVE * LDS_STRIDE];  // 33.8 KB

  const int tid   = threadIdx.x;
  const int wave  = tid >> 5;
  const int lane  = tid & 31;
  const int half  = lane >> 4;   // 0: K={0,1} / rows 0-7 of D; 1: K={2,3} / rows 8-15
  const int lh    = lane & 15;   // M index for A, N index for B/D
  const int slice = blockIdx.y;
  const int cbase = slice * DHALF;
  const int rowbase = blockIdx.x * ROWS_PER_BLOCK + wave * ROWS_PER_WAVE;

  // Stage this slice's weight chunk into LDS (read broadcast later).
  for (int i = tid * 4; i < DHALF; i += BLOCK_THREADS * 4) {
    *(v4f*)&sw[i] = *(const v4f*)&w[cbase + i];
  }

  // Issue one chunk's worth (16 rows x 64 cols = 8 x b128-per-lane) of
  // async global->LDS copies into buffer `buf`. Per-lane LDS dest keeps
  // the padded tile layout. Tracked by ASYNCcnt (8 per chunk).
  auto issue_chunk = [&](int c, int buf) {
    const int cg = cbase + c * KCHUNK;
    #pragma unroll
    for (int rr = 0; rr < 8; ++rr) {
      const int r = 2 * rr + half;  // each half-wave moves one contiguous 256B row segment
      const float* gsrc = G + (size_t)(rowbase + r) * DCOLS + cg + lh * 4;
      uint32_t loff = (uint32_t)(uintptr_t)&tile[buf][wave][r * LDS_STRIDE + lh * 4];
      asm volatile("global_load_async_to_lds_b128 %0, %1, off"
                   :: "v"(loff), "v"(gsrc)
                   : "memory");
    }
  };

  __syncthreads();  // sw visible to all waves; waves run decoupled after this

  v8f acc = {};
  issue_chunk(0, 0);

  for (int c = 0; c < NCHUNK; ++c) {
    const int buf = c & 1;
    if (c + 1 < NCHUNK) {
      issue_chunk(c + 1, buf ^ 1);
      if (c + 2 < NCHUNK) {  // warm L2 two chunks ahead (global_prefetch_b8)
        __builtin_prefetch(
            &G[(size_t)(rowbase + lh) * DCOLS + cbase + (c + 2) * KCHUNK + half * 32],
            0, 0);
      }
      // 8 in flight belong to chunk c+1; chunk c has landed in LDS.
      asm volatile("s_wait_asynccnt 0x8" ::: "memory");
    } else {
      asm volatile("s_wait_asynccnt 0x0" ::: "memory");
    }

    // 16 chained WMMA steps of K=4 over the 64-wide chunk.
    #pragma unroll
    for (int kk = 0; kk < KCHUNK; kk += 4) {
      // A fragment: lane lh holds row lh, K pair selected by half (K = 2*half+{0,1}).
      v2f a = *(const v2f*)&tile[buf][wave][lh * LDS_STRIDE + kk + 2 * half];
      // B fragment: B[k][n] = w[c*KCHUNK+kk+k] for all n (broadcast across columns).
      v2f b = *(const v2f*)&sw[c * KCHUNK + kk + 2 * half];
      acc = __builtin_amdgcn_wmma_f32_16x16x4_f32(
          /*neg_a=*/false, a, /*neg_b=*/false, b,
          /*c_mod=*/(short)0, acc, /*reuse_a=*/false, /*reuse_b=*/false);
    }
  }

  // D layout: VGPR r, lanes 0-15 -> M=r; lanes 16-31 -> M=8+r; all N identical.
  if (lh == 0) {
    float* dst = logits_partial + (size_t)slice * SROWS + rowbase + half * 8;
    #pragma unroll
    for (int r = 0; r < 8; ++r) dst[r] = acc[r];
  }
}

// One block per head: sum the K-slice partials, softmax over 8192 logits.
__global__ __launch_bounds__(1024) void softmax_kernel(
    const float* __restrict__ ws /* [2 heads][KSLICES][SROWS] */,
    float* __restrict__ out /* [2 heads][SROWS] */) {
  __shared__ float red[1024];
  const int head = blockIdx.x;
  const int tid  = threadIdx.x;
  const float* x0 = ws + (size_t)head * KSLICES * SROWS;
  const float* x1 = x0 + SROWS;
  float* y = out + (size_t)head * SROWS;

  float m = -INFINITY;
  for (int i = tid; i < SROWS; i += 1024) m = fmaxf(m, x0[i] + x1[i]);
  red[tid] = m;
  __syncthreads();
  for (int s = 512; s > 0; s >>= 1) {
    if (tid < s) red[tid] = fmaxf(red[tid], red[tid + s]);
    __syncthreads();
  }
  m = red[0];
  __syncthreads();

  float sum = 0.f;
  for (int i = tid; i < SROWS; i += 1024) sum += __expf(x0[i] + x1[i] - m);
  red[tid] = sum;
  __syncthreads();
  for (int s = 512; s > 0; s >>= 1) {
    if (tid < s) red[tid] += red[tid + s];
    __syncthreads();
  }
  const float inv = 1.0f / red[0];

  for (int i = tid; i < SROWS; i += 1024) y[i] = __expf(x0[i] + x1[i] - m) * inv;
}

extern "C" void kernel_launch(void* const* d_in, const int* in_sizes, int n_in,
                              void* d_out, int out_size, void* d_ws, size_t ws_size,
                              hipStream_t stream) {
  (void)in_sizes; (void)n_in; (void)out_size; (void)ws_size;
  const float* G1 = (const float*)d_in[0];
  const float* G2 = (const float*)d_in[1];
  const float* w1 = (const float*)d_in[2];
  const float* w2 = (const float*)d_in[3];
  float* out = (float*)d_out;
  float* ws  = (float*)d_ws;   // [2 heads][KSLICES][SROWS] partial logits = 128 KB

  dim3 grid(SROWS / ROWS_PER_BLOCK, KSLICES);  // (128, 2)
  gemv_wmma_kernel<<<grid, BLOCK_THREADS, 0, stream>>>(G1, w1, ws);
  gemv_wmma_kernel<<<grid, BLOCK_THREADS, 0, stream>>>(G2, w2, ws + (size_t)KSLICES * SROWS);
  softmax_kernel<<<2, 1024, 0, stream>>>(ws, out);
}